// WGATdAgg_62354335203996
// MI455X (gfx1250) — compile-verified
//
#include <hip/hip_runtime.h>
#include <hip/hip_bf16.h>

typedef __attribute__((ext_vector_type(2))) float v2f;
typedef __attribute__((ext_vector_type(8))) float v8f;

#define Bb 128
#define Ll 200
#define Dd 64
#define Hh 8
#define Ee 512
#define NEG_MASK_F (-2147483647.0f)

// ---------------- Kernel 1: e[b,l,n] = sum_d X[b,l,d]*W[n,d] + bias[n] ----------------
// M = B*L = 25600 rows, N = 512, K = 64.  One 16x16 tile per wave, f32 WMMA 16x16x4.
__global__ __launch_bounds__(256) void k_proj(const float* __restrict__ X,
                                              const float* __restrict__ W,
                                              const float* __restrict__ bias,
                                              float* __restrict__ e) {
  const int lane = threadIdx.x & 31;
  const int wave = threadIdx.x >> 5;
  const int tile = blockIdx.x * 8 + wave;          // 6400 blocks * 8 waves = 51200 tiles
  const int m0 = (tile >> 5) << 4;                 // 32 n-tiles per m-row
  const int n0 = (tile & 31) << 4;
  const int m = lane & 15;
  const int koff = (lane >> 4) << 1;               // lanes 16-31 hold K=+2,+3
  v8f acc = {0.f, 0.f, 0.f, 0.f, 0.f, 0.f, 0.f, 0.f};
  const float* xrow = X + (size_t)(m0 + m) * 64;   // A row for this lane
  const float* wrow = W + (size_t)(n0 + m) * 64;   // B col n = lane&15 ; B[k][n]=W[n*64+k]
  #pragma unroll
  for (int kk = 0; kk < 16; ++kk) {
    const int k4 = kk * 4;
    v2f a, bb;
    a.x = xrow[k4 + koff];
    a.y = xrow[k4 + koff + 1];
    bb.x = wrow[k4 + koff];
    bb.y = wrow[k4 + koff + 1];
    acc = __builtin_amdgcn_wmma_f32_16x16x4_f32(false, a, false, bb, (short)0, acc,
                                                false, false);
  }
  const int col = n0 + m;
  const float bv = bias[col];
  #pragma unroll
  for (int r = 0; r < 8; ++r) {
    const int row = m0 + r + ((lane >> 4) << 3);   // C/D layout: VGPR r -> M=r / r+8
    e[(size_t)row * Ee + col] = acc[r] + bv;
  }
}

// ---------------- Kernel 2: p1[b,h,l], p2[b,h,l] = sum_d e[b,l,h*64+d]*att{1,2}[h*64+d]
__global__ void k_p12(const float* __restrict__ e, const float* __restrict__ att1,
                      const float* __restrict__ att2, float* __restrict__ p1,
                      float* __restrict__ p2) {
  const int idx = blockIdx.x * blockDim.x + threadIdx.x;
  if (idx >= Bb * Hh * Ll) return;
  const int b = idx / (Hh * Ll);
  const int rem = idx % (Hh * Ll);
  const int h = rem / Ll;
  const int l = rem % Ll;
  const float* ev = e + (size_t)b * Ll * Ee + (size_t)l * Ee + h * 64;
  const float* a1 = att1 + h * 64;
  const float* a2 = att2 + h * 64;
  float s1 = 0.f, s2 = 0.f;
  #pragma unroll 8
  for (int d = 0; d < 64; ++d) {
    const float v = ev[d];
    s1 += v * a1[d];
    s2 += v * a2[d];
  }
  p1[idx] = s1;
  p2[idx] = s2;
}

// ---------------- Kernel 3: per (b,h): column softmax + alpha @ ev ----------------
// weighted[b,h,i,d] = sum_j alpha[b,h,i,j] * ev[b,h,j,d]
// alpha softmax is over i (axis=2) -> per-column (j) max & denom.
__global__ __launch_bounds__(256) void k_attn(const float* __restrict__ e,
                                              const float* __restrict__ adj,
                                              const float* __restrict__ att3,
                                              const float* __restrict__ p1g,
                                              const float* __restrict__ p2g,
                                              float* __restrict__ wt) {
  __shared__ float p1s[Ll];
  __shared__ float p2s[Ll];
  __shared__ float cmax[Ll];
  __shared__ float rden[Ll];
  __shared__ float alpha_s[208 * 17];              // 16-wide K panel, odd stride (no conflicts)

  const int bh = blockIdx.x;                       // 1024 = B*H
  const int b = bh >> 3, h = bh & 7;
  const int tid = threadIdx.x;
  const int lane = tid & 31, wave = tid >> 5;      // wave32
  const float a3 = att3[h];
  const float* adjb = adj + (size_t)b * (Ll * Ll);
  const float* evb = e + (size_t)b * (Ll * Ee) + (size_t)h * (Ll * 64); // raw-reshape view

  if (tid < Ll) {
    p1s[tid] = p1g[bh * Ll + tid];
    p2s[tid] = p2g[bh * Ll + tid];
  }
  __syncthreads();

  // ---- column-wise (over i) max and softmax denominator; thread = column j ----
  if (tid < Ll) {
    const float p2v = p2s[tid];
    float mx = -3.0e38f;
    for (int i = 0; i < Ll; ++i) {
      const float a = adjb[(size_t)i * Ll + tid];  // lane-contiguous j: coalesced
      float t = p1s[i] + p2v + a3 * a;
      t = t >= 0.f ? t : 0.2f * t;                 // leaky_relu(0.2)
      if (a == 0.f) t += NEG_MASK_F;               // mask
      mx = fmaxf(mx, t);
    }
    float den = 0.f;
    for (int i = 0; i < Ll; ++i) {
      const float a = adjb[(size_t)i * Ll + tid];
      float t = p1s[i] + p2v + a3 * a;
      t = t >= 0.f ? t : 0.2f * t;
      if (a == 0.f) t += NEG_MASK_F;
      den += __expf(t - mx);
    }
    cmax[tid] = mx;
    rden[tid] = 1.0f / den;
  }
  __syncthreads();

  // ---- GEMM: M=200 (13 i-tiles, zero-padded), N=64 (4 d-tiles), K=200 (13 panels) ----
  const v8f vzero = {0.f, 0.f, 0.f, 0.f, 0.f, 0.f, 0.f, 0.f};
  v8f acc[7];
  #pragma unroll
  for (int s = 0; s < 7; ++s) acc[s] = vzero;

  const int m = lane & 15;
  const int koff = (lane >> 4) << 1;

  for (int kb = 0; kb < 13; ++kb) {
    const int j0 = kb * 16;
    // cooperatively build alpha panel [208 x 16] (rows/cols >=200 are zero)
    #pragma unroll 1
    for (int it = 0; it < 13; ++it) {
      const int flat = it * 256 + tid;             // 0 .. 3327
      const int i = flat >> 4;
      const int jj = flat & 15;
      const int j = j0 + jj;
      float val = 0.f;
      if (i < Ll && j < Ll) {
        const float a = adjb[(size_t)i * Ll + j];
        float t = p1s[i] + p2s[j] + a3 * a;
        t = t >= 0.f ? t : 0.2f * t;
        if (a == 0.f) t += NEG_MASK_F;
        val = __expf(t - cmax[j]) * rden[j];
      }
      alpha_s[i * 17 + jj] = val;
    }
    __syncthreads();

    #pragma unroll
    for (int s = 0; s < 7; ++s) {
      const int t = wave + s * 8;                  // wave-uniform -> EXEC stays all-ones
      if (t < 52) {
        const int i0 = (t >> 2) << 4;
        const int d0 = (t & 3) << 4;
        #pragma unroll
        for (int kk = 0; kk < 4; ++kk) {
          const int kbase = kk * 4 + koff;
          v2f a, bb;
          a.x = alpha_s[(i0 + m) * 17 + kbase];
          a.y = alpha_s[(i0 + m) * 17 + kbase + 1];
          int r0 = j0 + kbase;
          int r1 = r0 + 1;
          r0 = r0 > (Ll - 1) ? (Ll - 1) : r0;      // clamp addr; alpha=0 kills contribution
          r1 = r1 > (Ll - 1) ? (Ll - 1) : r1;
          bb.x = evb[(size_t)r0 * 64 + d0 + m];
          bb.y = evb[(size_t)r1 * 64 + d0 + m];
          acc[s] = __builtin_amdgcn_wmma_f32_16x16x4_f32(false, a, false, bb, (short)0,
                                                         acc[s], false, false);
        }
      }
    }
    __syncthreads();
  }

  // ---- store weighted[b,h] tiles ----
  float* wtb = wt + (size_t)bh * (Ll * 64);
  #pragma unroll
  for (int s = 0; s < 7; ++s) {
    const int t = wave + s * 8;
    if (t < 52) {
      const int i0 = (t >> 2) << 4;
      const int d0 = (t & 3) << 4;
      #pragma unroll
      for (int r = 0; r < 8; ++r) {
        const int row = i0 + r + ((lane >> 4) << 3);
        if (row < Ll) wtb[(size_t)row * 64 + d0 + m] = acc[s][r];
      }
    }
  }
}

// ---------------- Kernel 4: out[b,l,d] = relu(mean_h weighted[b,h,l,d]) ----------------
__global__ void k_out(const float* __restrict__ wt, float* __restrict__ out) {
  const int idx = blockIdx.x * blockDim.x + threadIdx.x;
  if (idx >= Bb * Ll * Dd) return;
  const int b = idx / (Ll * Dd);
  const int rem = idx % (Ll * Dd);
  const float* base = wt + (size_t)b * (Hh * Ll * Dd) + rem;
  float s = 0.f;
  #pragma unroll
  for (int h = 0; h < Hh; ++h) s += base[h * (Ll * Dd)];
  s *= 0.125f;
  out[idx] = s > 0.f ? s : 0.f;
}

extern "C" void kernel_launch(void* const* d_in, const int* in_sizes, int n_in,
                              void* d_out, int out_size, void* d_ws, size_t ws_size,
                              hipStream_t stream) {
  const float* X    = (const float*)d_in[0];   // item_seq_emb [B,L,D]
  // d_in[1] = adj_in : unused by the reference computation
  const float* adjo = (const float*)d_in[2];   // adj_out [B,L,L]
  const float* W    = (const float*)d_in[3];   // [512,64]
  const float* bias = (const float*)d_in[4];   // [512]
  const float* att1 = (const float*)d_in[5];   // [1,512]
  const float* att2 = (const float*)d_in[6];   // [1,512]
  const float* att3 = (const float*)d_in[7];   // [1,8]
  float* out = (float*)d_out;

  float* ws = (float*)d_ws;
  float* e  = ws;                              // B*L*E      = 13,107,200 f
  float* p1 = e + (size_t)Bb * Ll * Ee;        // B*H*L      =    204,800 f
  float* p2 = p1 + (size_t)Bb * Hh * Ll;       //    204,800 f
  float* wt = p2 + (size_t)Bb * Hh * Ll;       // B*H*L*D    = 13,107,200 f

  k_proj<<<6400, 256, 0, stream>>>(X, W, bias, e);
  k_p12<<<(Bb * Hh * Ll + 255) / 256, 256, 0, stream>>>(e, att1, att2, p1, p2);
  k_attn<<<Bb * Hh, 256, 0, stream>>>(e, adjo, att3, p1, p2, wt);
  k_out<<<(Bb * Ll * Dd + 255) / 256, 256, 0, stream>>>(wt, out);
}